// GTransformer_31533649887968
// MI455X (gfx1250) — compile-verified
//
#include <hip/hip_runtime.h>
#include <hip/hip_bf16.h>
#include <math.h>

// ---------------- problem constants (match reference) ----------------
#define N_NODES 4096
#define DIN     256
#define H1DIM   128
#define H2DIM   64
#define NHEADS  2
#define ALPHAF  0.2f
#define NEGF    (-9.0e15f)

typedef float v2f __attribute__((ext_vector_type(2)));
typedef float v8f __attribute__((ext_vector_type(8)));
typedef unsigned int u32x4 __attribute__((ext_vector_type(4)));
typedef int i32x8 __attribute__((ext_vector_type(8)));
typedef int i32x4 __attribute__((ext_vector_type(4)));

__device__ __forceinline__ float lrelu(float x) {
    return x > 0.0f ? x : ALPHAF * x;
}

#define WMMA_F32(a, b, c) \
    __builtin_amdgcn_wmma_f32_16x16x4_f32(false, (a), false, (b), (short)0, (c), false, false)

#if __has_builtin(__builtin_amdgcn_tensor_load_to_lds)
#define HAVE_TDM 1
#else
#define HAVE_TDM 0
#endif

// Low 32 bits of a generic pointer to LDS == the LDS byte offset
// (generic LDS addr = {shared_aperture_hi32, lds_offset32}).
__device__ __forceinline__ unsigned lds_off_u32(const void* p) {
    return (unsigned)(size_t)p;
}

#if HAVE_TDM
// Issue a 2D TDM load: `rows` x `ncols` f32 tile from row-major global
// (row stride = ncols elements) into LDS at lds_off, padding the LDS
// destination by 8 DWORDs after every row of `ncols` DWORDs
// (LDS row stride becomes ncols+8 floats -> conflict-free ds_loads).
// D# layout per CDNA5 ISA sec 8.3/8.4. Group2/3 disabled (2D tensor).
// Toolchain uses the 6-arg builtin: (g0, g1, g2, g3, g_extra, cpol).
__device__ __forceinline__ void tdm_load_2d(unsigned lds_off, const float* gsrc,
                                            int rows, int ncols,
                                            int pad_interval_code) {
    unsigned long long ga = (unsigned long long)(size_t)gsrc;
    u32x4 g0;
    g0[0] = 1u;                                   // count=1, user mode, no gather
    g0[1] = lds_off;                              // lds_addr
    g0[2] = (unsigned)(ga & 0xffffffffull);       // global_addr[31:0]
    g0[3] = (unsigned)((ga >> 32) & 0x01ffffffull) | 0x80000000u; // [56:32] | type=2
    i32x8 g1;
    g1[0] = (2 << 16)        // data_size = 4 bytes
          | (1 << 20)        // pad_enable
          | (pad_interval_code << 22)   // pad every 2^(code+1) DWORDs (== ncols)
          | (7 << 25);       // pad_amount code 7 -> 8 DWORDs
    g1[1] = (ncols & 0xffff) << 16;               // tensor_dim0[15:0]
    g1[2] = ((ncols >> 16) & 0xffff)              // tensor_dim0[31:16]
          | ((N_NODES & 0xffff) << 16);           // tensor_dim1[15:0]
    g1[3] = ((N_NODES >> 16) & 0xffff)            // tensor_dim1[31:16]
          | ((ncols & 0xffff) << 16);             // tile_dim0
    g1[4] = rows & 0xffff;                        // tile_dim1 (tile_dim2 = 0)
    g1[5] = ncols;                                // tensor_dim0_stride[31:0]
    g1[6] = 0;                                    // stride hi / dim1_stride lo
    g1[7] = 0;
    i32x4 z4 = {0, 0, 0, 0};
    i32x8 z8 = {0, 0, 0, 0, 0, 0, 0, 0};
    __builtin_amdgcn_tensor_load_to_lds(g0, g1, z4, z4, z8, 0);
}
#endif

// =====================================================================
// Kernel 1: per-head GEMM  H[head] = A(4096 x K) @ W[head](K x Nc)
// One wave computes a 16-row x Nc strip; f32 WMMA 16x16x4, K-step 4.
// (0.5 GFLOP total -- not the hot path.)
// =====================================================================
template <int NT>
__global__ void gemm_strip(const float* __restrict__ A,
                           const float* __restrict__ W,
                           float* __restrict__ H,
                           int K) {
    const int Nc   = NT * 16;
    const int head = blockIdx.y;
    const int row0 = blockIdx.x * 16;
    const int lane = threadIdx.x;
    const int lm   = lane & 15;
    const int half = lane >> 4;

    const float* Wh = W + (size_t)head * K * Nc;
    float*       Hh = H + (size_t)head * N_NODES * Nc;

    v8f acc[NT];
#pragma unroll
    for (int t = 0; t < NT; ++t) acc[t] = {};

    const float* Arow = A + (size_t)(row0 + lm) * K;

    // software-pipelined B fragments (one K-step ahead)
    v2f bcur[NT], bnxt[NT];
    {
        const int kk = half * 2;
#pragma unroll
        for (int t = 0; t < NT; ++t) {
            bcur[t].x = Wh[(size_t)kk * Nc + t * 16 + lm];
            bcur[t].y = Wh[(size_t)(kk + 1) * Nc + t * 16 + lm];
        }
    }
    for (int k = 0; k < K; k += 4) {
        const int kk = k + half * 2;
        const int kn = (k + 4 < K) ? kk + 4 : half * 2;
#pragma unroll
        for (int t = 0; t < NT; ++t) {
            bnxt[t].x = Wh[(size_t)kn * Nc + t * 16 + lm];
            bnxt[t].y = Wh[(size_t)(kn + 1) * Nc + t * 16 + lm];
        }
        v2f a = *(const v2f*)(Arow + kk);
#pragma unroll
        for (int t = 0; t < NT; ++t) {
            acc[t] = WMMA_F32(a, bcur[t], acc[t]);
            bcur[t] = bnxt[t];
        }
    }

#pragma unroll
    for (int t = 0; t < NT; ++t)
#pragma unroll
        for (int v = 0; v < 8; ++v) {
            const int row = row0 + v + half * 8;
            Hh[(size_t)row * Nc + t * 16 + lm] = acc[t][v];
        }
}

// =====================================================================
// Kernel 2: wh1_i = h_i . a[0:d],  wh2_i = h_i . a[d:2d]   (wave per row)
// =====================================================================
__global__ void wh_kernel(const float* __restrict__ Hmat,
                          const float* __restrict__ avec,
                          float* __restrict__ wh1v,
                          float* __restrict__ wh2v,
                          int d) {
    const int head = blockIdx.y;
    const int wave = threadIdx.x >> 5;
    const int lane = threadIdx.x & 31;
    const int row  = blockIdx.x * 8 + wave;

    const float* h = Hmat + ((size_t)head * N_NODES + row) * d;
    const float* a = avec + (size_t)head * 2 * d;

    float s1 = 0.0f, s2 = 0.0f;
    for (int j = lane; j < d; j += 32) {
        const float hv = h[j];
        s1 += hv * a[j];
        s2 += hv * a[d + j];
    }
#pragma unroll
    for (int m = 16; m > 0; m >>= 1) {
        s1 += __shfl_xor(s1, m, 32);
        s2 += __shfl_xor(s2, m, 32);
    }
    if (lane == 0) {
        wh1v[head * N_NODES + row] = s1;
        wh2v[head * N_NODES + row] = s2;
    }
}

// =====================================================================
// Kernel 3: exact two-pass masked softmax row statistics (wave per row)
// =====================================================================
__global__ void row_stats(const int* __restrict__ adj,
                          const float* __restrict__ wh1v,
                          const float* __restrict__ wh2v,
                          float* __restrict__ mrow,
                          float* __restrict__ srowinv) {
    const int head = blockIdx.y;
    const int wave = threadIdx.x >> 5;
    const int lane = threadIdx.x & 31;
    const int row  = blockIdx.x * 8 + wave;

    const float* w2 = wh2v + (size_t)head * N_NODES;
    const float  w1 = wh1v[head * N_NODES + row];
    const int*   ar = adj + (size_t)row * N_NODES;

    float m = -3.0e38f;
    for (int j = lane; j < N_NODES; j += 32) {
        const float v = (ar[j] > 0) ? lrelu(w1 + w2[j]) : NEGF;
        m = fmaxf(m, v);
    }
#pragma unroll
    for (int d = 16; d > 0; d >>= 1) m = fmaxf(m, __shfl_xor(m, d, 32));

    float s = 0.0f;
    for (int j = lane; j < N_NODES; j += 32) {
        const float v = (ar[j] > 0) ? lrelu(w1 + w2[j]) : NEGF;
        s += __expf(v - m);   // masked terms underflow to 0 (matches reference)
    }
#pragma unroll
    for (int d = 16; d > 0; d >>= 1) s += __shfl_xor(s, d, 32);

    if (lane == 0) {
        mrow[head * N_NODES + row]    = m;
        srowinv[head * N_NODES + row] = 1.0f / s;
    }
}

// =====================================================================
// Kernel 4: fused  O = normalize(lrelu(att @ H)) + bias  (per head)
// Block = 4 waves = 64 rows. H is staged in 32 x Nc K-chunks into LDS
// by the Tensor Data Mover (double-buffered, issued by wave 0, waited
// with s_wait_tensorcnt, released via workgroup barrier). Attention
// A-fragments are recomputed on-the-fly (rank-1 logits + adj mask),
// with the global adj/wh2 reads software-pipelined one K-step ahead.
// =====================================================================
template <int NT>
__global__ void __launch_bounds__(128)
attout_tdm(const int* __restrict__ adj,
           const float* __restrict__ wh1v,
           const float* __restrict__ wh2v,
           const float* __restrict__ mrow,
           const float* __restrict__ srowinv,
           const float* __restrict__ Hmat,
           const float* __restrict__ bias,
           float* __restrict__ Out) {
    constexpr int Nc = NT * 16;
    constexpr int KT = 32;            // K-chunk (columns of att / rows of H)
    constexpr int LS = Nc + 8;        // LDS row stride (floats): pad kills conflicts
    constexpr int NCHUNK = N_NODES / KT;
    // pad interval code: pad after 2^(code+1) DWORDs == Nc
    constexpr int PADC = (Nc == 128) ? 6 : 5;

    __shared__ float lbuf[2][KT * LS];

    const int head = blockIdx.y;
    const int wave = threadIdx.x >> 5;
    const int lane = threadIdx.x & 31;
    const int lm   = lane & 15;
    const int half = lane >> 4;
    const int row0 = (blockIdx.x * 4 + wave) * 16;

    const float* Hh  = Hmat + (size_t)head * N_NODES * Nc;
    const float* w2h = wh2v + (size_t)head * N_NODES;
    const int    irw = row0 + lm;
    const float  w1  = wh1v[head * N_NODES + irw];
    const float  mi  = mrow[head * N_NODES + irw];
    const float  si  = srowinv[head * N_NODES + irw];
    const int*   arj = adj + (size_t)irw * N_NODES;

    v8f acc[NT];
#pragma unroll
    for (int t = 0; t < NT; ++t) acc[t] = {};

#if HAVE_TDM
    if (wave == 0) tdm_load_2d(lds_off_u32(&lbuf[0][0]), Hh, KT, Nc, PADC);
#endif

    // adj / wh2 software pipeline (one K-step ahead)
    int2 av  = *(const int2*)(arj + half * 2);
    v2f  w2v = *(const v2f*)(w2h + half * 2);

    for (int c = 0; c < NCHUNK; ++c) {
        const int cur = c & 1;
#if HAVE_TDM
        if (wave == 0) __builtin_amdgcn_s_wait_tensorcnt(0);
        __syncthreads();                        // chunk c visible to all waves
        if (wave == 0 && (c + 1) < NCHUNK)      // overlap next DMA with compute
            tdm_load_2d(lds_off_u32(&lbuf[cur ^ 1][0]),
                        Hh + (size_t)(c + 1) * KT * Nc, KT, Nc, PADC);
#else
        // fallback: cooperative staging with vector loads
        __syncthreads();
        {
            const float* src = Hh + (size_t)c * KT * Nc;
            for (int idx = threadIdx.x; idx < KT * Nc; idx += 128) {
                const int r = idx / Nc, col = idx - r * Nc;
                lbuf[cur][r * LS + col] = src[idx];
            }
        }
        __syncthreads();
#endif
        const float* lb = lbuf[cur];
#pragma unroll 4
        for (int kk = 0; kk < KT; kk += 4) {
            // prefetch next step's adj / wh2 (wraps harmlessly at the end)
            const int jg = c * KT + kk + half * 2;
            const int jn = (jg + 4 < N_NODES) ? jg + 4 : half * 2;
            const int2 avn = *(const int2*)(arj + jn);
            const v2f  w2n = *(const v2f*)(w2h + jn);

            // attention A-fragment for this K-step
            const float v0 = (av.x > 0) ? lrelu(w1 + w2v.x) : NEGF;
            const float v1 = (av.y > 0) ? lrelu(w1 + w2v.y) : NEGF;
            v2f a;
            a.x = __expf(v0 - mi) * si;
            a.y = __expf(v1 - mi) * si;

            const float* lrow = lb + (size_t)(kk + half * 2) * LS;
#pragma unroll
            for (int t = 0; t < NT; ++t) {
                v2f b;
                b.x = lrow[t * 16 + lm];          // row kk+half*2
                b.y = lrow[LS + t * 16 + lm];     // row kk+half*2+1
                acc[t] = WMMA_F32(a, b, acc[t]);
            }
            av = avn;
            w2v = w2n;
        }
        __syncthreads();                          // done reading lbuf[cur]
    }

    // leaky-relu, per-row L2 norm (in-register via 16-lane shfl_xor), bias
    float q[8];
#pragma unroll
    for (int v = 0; v < 8; ++v) q[v] = 0.0f;
#pragma unroll
    for (int t = 0; t < NT; ++t)
#pragma unroll
        for (int v = 0; v < 8; ++v) {
            float x = acc[t][v];
            x = x > 0.0f ? x : ALPHAF * x;
            acc[t][v] = x;
            q[v] += x * x;
        }
#pragma unroll
    for (int v = 0; v < 8; ++v) {
        q[v] += __shfl_xor(q[v], 1, 16);
        q[v] += __shfl_xor(q[v], 2, 16);
        q[v] += __shfl_xor(q[v], 4, 16);
        q[v] += __shfl_xor(q[v], 8, 16);
    }

    float* Oh = Out + (size_t)head * N_NODES * Nc;
#pragma unroll
    for (int t = 0; t < NT; ++t) {
        const float bl = bias[head * Nc + t * 16 + lm];
#pragma unroll
        for (int v = 0; v < 8; ++v) {
            const float nrm = fmaxf(sqrtf(q[v]), 1e-12f);
            const int   row = row0 + v + half * 8;
            Oh[(size_t)row * Nc + t * 16 + lm] = acc[t][v] / nrm + bl;
        }
    }
}

// =====================================================================
// Head-mean combiners
// =====================================================================
__global__ void combine_elu(const float* __restrict__ O, float* __restrict__ X, int total) {
    const int i = blockIdx.x * blockDim.x + threadIdx.x;
    if (i < total) {
        const float z = 0.5f * (O[i] + O[total + i]);
        X[i] = z > 0.0f ? z : (__expf(z) - 1.0f);   // elu, alpha=1
    }
}

__global__ void combine_mean(const float* __restrict__ O, float* __restrict__ X, int total) {
    const int i = blockIdx.x * blockDim.x + threadIdx.x;
    if (i < total) X[i] = 0.5f * (O[i] + O[total + i]);
}

// =====================================================================
// Launch
// =====================================================================
extern "C" void kernel_launch(void* const* d_in, const int* in_sizes, int n_in,
                              void* d_out, int out_size, void* d_ws, size_t ws_size,
                              hipStream_t stream) {
    const float* x   = (const float*)d_in[0];   // [4096,256]
    const int*   adj = (const int*)  d_in[1];   // [4096,4096]
    const float* W1  = (const float*)d_in[2];   // [2,256,128]
    const float* a1  = (const float*)d_in[3];   // [2,256,1]
    const float* b1  = (const float*)d_in[4];   // [2,128]
    const float* W2  = (const float*)d_in[5];   // [2,128,64]
    const float* a2  = (const float*)d_in[6];   // [2,128,1]
    const float* b2  = (const float*)d_in[7];   // [2,64]
    float* out = (float*)d_out;                 // [4096,64]
    float* ws  = (float*)d_ws;

    float* h1  = ws;                                    // 2*4096*128
    float* o1  = h1 + (size_t)NHEADS * N_NODES * H1DIM; // 2*4096*128
    float* x1  = o1 + (size_t)NHEADS * N_NODES * H1DIM; // 4096*128
    float* h2  = x1 + (size_t)N_NODES * H1DIM;          // 2*4096*64
    float* o2  = h2 + (size_t)NHEADS * N_NODES * H2DIM; // 2*4096*64
    float* vec = o2 + (size_t)NHEADS * N_NODES * H2DIM; // 8*4096 (reused per layer)
    float* wh1v = vec;
    float* wh2v = vec + (size_t)NHEADS * N_NODES;
    float* mrow = vec + 2 * (size_t)NHEADS * N_NODES;
    float* srow = vec + 3 * (size_t)NHEADS * N_NODES;

    const dim3 gStrip(N_NODES / 16, NHEADS);     // 1 wave per 16-row strip
    const dim3 gQuad(N_NODES / 64, NHEADS);      // 4 waves (64 rows) per block
    const dim3 gRow(N_NODES / 8, NHEADS);        // 8 rows per 256-thread block

    // ---- Layer 1 (d = 128) ----
    gemm_strip<8><<<gStrip, 32, 0, stream>>>(x, W1, h1, DIN);
    wh_kernel  <<<gRow, 256, 0, stream>>>(h1, a1, wh1v, wh2v, H1DIM);
    row_stats  <<<gRow, 256, 0, stream>>>(adj, wh1v, wh2v, mrow, srow);
    attout_tdm<8><<<gQuad, 128, 0, stream>>>(adj, wh1v, wh2v, mrow, srow, h1, b1, o1);
    combine_elu<<<(N_NODES * H1DIM + 255) / 256, 256, 0, stream>>>(o1, x1, N_NODES * H1DIM);

    // ---- Layer 2 (d = 64) ----
    gemm_strip<4><<<gStrip, 32, 0, stream>>>(x1, W2, h2, H1DIM);
    wh_kernel  <<<gRow, 256, 0, stream>>>(h2, a2, wh1v, wh2v, H2DIM);
    row_stats  <<<gRow, 256, 0, stream>>>(adj, wh1v, wh2v, mrow, srow);
    attout_tdm<4><<<gQuad, 128, 0, stream>>>(adj, wh1v, wh2v, mrow, srow, h2, b2, o2);
    combine_mean<<<(N_NODES * H2DIM + 255) / 256, 256, 0, stream>>>(o2, out, N_NODES * H2DIM);
}